// ENCODER_1185410974359
// MI455X (gfx1250) — compile-verified
//
#include <hip/hip_runtime.h>

// GraphSAGE VGAE encoder for MI455X (gfx1250), fp32 end-to-end.
// Dense layers use V_WMMA_F32_16X16X4_F32 (fp32 matrix cores, exact precision).
// Aggregation is L2-resident scatter-add (atomics) — the actual bottleneck.

#define LN_EPS 1e-5f

typedef __attribute__((ext_vector_type(2))) float v2f;
typedef __attribute__((ext_vector_type(8))) float v8f;

// ---------------- utility kernels ----------------

__global__ void vgae_zero(float* __restrict__ p, long n) {
  long i = (long)blockIdx.x * blockDim.x + threadIdx.x;
  if (i < n) p[i] = 0.0f;
}

__global__ void vgae_deg(const int* __restrict__ ei, float* __restrict__ deg, int E) {
  int e = blockIdx.x * blockDim.x + threadIdx.x;
  if (e < E) atomicAdd(&deg[ei[E + e]], 1.0f);   // dst row of edge_index
}

__global__ void vgae_invdeg(float* __restrict__ d, int n) {
  int i = blockIdx.x * blockDim.x + threadIdx.x;
  if (i < n) d[i] = 1.0f / fmaxf(d[i], 1.0f);
}

// Mean-aggregation numerator: agg[dst,:] += h[src,:]  (scaled by inv_deg later,
// inside the GEMM A-fragment load). blockDim = (d, 256/d): lane = feature dim.
__global__ void vgae_agg(const float* __restrict__ h, const int* __restrict__ ei,
                         float* __restrict__ agg, int d, int E) {
  int e = blockIdx.x * blockDim.y + threadIdx.y;
  if (e >= E) return;
  int s = ei[e];        // src
  int t = ei[E + e];    // dst
  float v = h[(long)s * d + threadIdx.x];
  atomicAdd(&agg[(long)t * d + threadIdx.x], v);
}

// ---------------- fused SAGE dense layer ----------------
// out[16 x dout] tile per block:  out = (agg*inv_deg) @ Wl^T + bl + h @ Wr^T
// => single GEMM, K = 2*din, A = [agg*invd | h], B^T rows from Wl then Wr.
// wave w owns column tile w (dout/16 waves per block). Optional LN+ReLU epilogue.
//
// V_WMMA_F32_16X16X4_F32 fragment layout (wave32):
//   A 16x4:  lane L -> row (L&15), VGPR0/1 = K = k0+2*(L>=16) and +1
//   B 4x16:  lane L -> col (L&15), VGPR0/1 = K = k0+2*(L>=16) and +1
//   C 16x16: vgpr v, lane L -> row v+8*(L>=16), col (L&15)
__global__ void vgae_sage_gemm(const float* __restrict__ agg,
                               const float* __restrict__ h,
                               const float* __restrict__ invdeg,
                               const float* __restrict__ Wl,
                               const float* __restrict__ bl,
                               const float* __restrict__ Wr,
                               const float* __restrict__ gamma,
                               const float* __restrict__ beta,
                               float* __restrict__ out,
                               int din, int dout, int do_ln) {
  __shared__ float smem[16 * 128];
  __shared__ float s_mean[16];
  __shared__ float s_rstd[16];

  const int lane = threadIdx.x & 31;
  const int wave = threadIdx.x >> 5;
  const int l15  = lane & 15;
  const int hi   = lane >> 4;          // 0: lanes 0-15, 1: lanes 16-31
  const int rowbase = blockIdx.x * 16; // N is an exact multiple of 16 (50000)
  const int arow = rowbase + l15;
  const int bcol = wave * 16 + l15;
  const int koff = 2 * hi;

  const float id = invdeg[arow];
  const float* aggrow = agg + (long)arow * din;
  const float* hrow   = h   + (long)arow * din;
  const float* wlrow  = Wl  + (long)bcol * din;   // Wl is (dout, din) row-major
  const float* wrrow  = Wr  + (long)bcol * din;

  v8f c = {0.f, 0.f, 0.f, 0.f, 0.f, 0.f, 0.f, 0.f};

  // K-half 1: mean-aggregated neighbors through Wl
  for (int k = 0; k < din; k += 4) {
    v2f a = *(const v2f*)(aggrow + k + koff);
    a *= id;                                    // fold inv_deg into A fragment
    v2f b = *(const v2f*)(wlrow + k + koff);
    c = __builtin_amdgcn_wmma_f32_16x16x4_f32(false, a, false, b,
                                              (short)0, c, false, false);
  }
  // K-half 2: root features through Wr
  for (int k = 0; k < din; k += 4) {
    v2f a = *(const v2f*)(hrow + k + koff);
    v2f b = *(const v2f*)(wrrow + k + koff);
    c = __builtin_amdgcn_wmma_f32_16x16x4_f32(false, a, false, b,
                                              (short)0, c, false, false);
  }

  const float bias = bl[bcol];

  if (!do_ln) {                                 // final layer: store directly
    for (int v = 0; v < 8; ++v) {
      int r = rowbase + v + 8 * hi;
      out[(long)r * dout + bcol] = c[v] + bias;
    }
    return;
  }

  // LN + ReLU epilogue (dout == 128): stage full rows in LDS
  for (int v = 0; v < 8; ++v)
    smem[(v + 8 * hi) * 128 + bcol] = c[v] + bias;
  __syncthreads();

  if (threadIdx.x < 16) {
    const float* rowp = &smem[threadIdx.x * 128];
    float s = 0.f, ss = 0.f;
    for (int j = 0; j < 128; ++j) { float x = rowp[j]; s += x; ss += x * x; }
    float m   = s * (1.0f / 128.0f);
    float var = ss * (1.0f / 128.0f) - m * m;   // biased variance (matches jnp.var)
    s_mean[threadIdx.x] = m;
    s_rstd[threadIdx.x] = rsqrtf(var + LN_EPS);
  }
  __syncthreads();

  for (int idx = threadIdx.x; idx < 16 * 128; idx += blockDim.x) {
    int r  = idx >> 7;
    int cc = idx & 127;
    float x = (smem[idx] - s_mean[r]) * s_rstd[r] * gamma[cc] + beta[cc];
    out[(long)(rowbase + r) * 128 + cc] = fmaxf(x, 0.0f);
  }
}

// ---------------- launch ----------------

extern "C" void kernel_launch(void* const* d_in, const int* in_sizes, int n_in,
                              void* d_out, int out_size, void* d_ws, size_t ws_size,
                              hipStream_t stream) {
  const float* x  = (const float*)d_in[0];
  const int*   ei = (const int*)d_in[1];
  const int N = in_sizes[0] / 64;   // 50000
  const int E = in_sizes[1] / 2;    // 800000

  // Workspace layout (fp32): invdeg[N] | aggx[N*64] | bufA[N*128] | bufB[N*128] | bufC[N*128]
  float* invdeg = (float*)d_ws;
  float* aggx   = invdeg + N;
  float* bufA   = aggx + (long)N * 64;
  float* bufB   = bufA + (long)N * 128;
  float* bufC   = bufB + (long)N * 128;

  // degree -> inv_deg (shared by both towers)
  vgae_zero<<<(N + 255) / 256, 256, 0, stream>>>(invdeg, N);
  vgae_deg<<<(E + 255) / 256, 256, 0, stream>>>(ei, invdeg, E);
  vgae_invdeg<<<(N + 255) / 256, 256, 0, stream>>>(invdeg, N);

  // layer-0 aggregation of x: identical for both towers, compute once
  vgae_zero<<<((long)N * 64 + 255) / 256, 256, 0, stream>>>(aggx, (long)N * 64);
  {
    dim3 blk(64, 4);
    vgae_agg<<<(E + 3) / 4, blk, 0, stream>>>(x, ei, aggx, 64, E);
  }

  const int nrt = N / 16;  // 3125 row tiles, exact
  for (int t = 0; t < 2; ++t) {  // t=0: mu tower, t=1: logvar tower
    const int base = 2 + t * 13;
    const float* Wl0 = (const float*)d_in[base + 0];
    const float* bl0 = (const float*)d_in[base + 1];
    const float* Wr0 = (const float*)d_in[base + 2];
    const float* g0  = (const float*)d_in[base + 3];
    const float* b0  = (const float*)d_in[base + 4];
    const float* Wl1 = (const float*)d_in[base + 5];
    const float* bl1 = (const float*)d_in[base + 6];
    const float* Wr1 = (const float*)d_in[base + 7];
    const float* g1  = (const float*)d_in[base + 8];
    const float* b1  = (const float*)d_in[base + 9];
    const float* Wl2 = (const float*)d_in[base + 10];
    const float* bl2 = (const float*)d_in[base + 11];
    const float* Wr2 = (const float*)d_in[base + 12];
    float* outp = (float*)d_out + (long)t * N * 64;

    // layer 0: (aggx, x) -> h1 = bufA   [64 -> 128, LN+ReLU]
    vgae_sage_gemm<<<nrt, 256, 0, stream>>>(aggx, x, invdeg, Wl0, bl0, Wr0,
                                            g0, b0, bufA, 64, 128, 1);
    // agg(h1) -> bufB
    vgae_zero<<<((long)N * 128 + 255) / 256, 256, 0, stream>>>(bufB, (long)N * 128);
    { dim3 blk(128, 2); vgae_agg<<<(E + 1) / 2, blk, 0, stream>>>(bufA, ei, bufB, 128, E); }

    // layer 1: (bufB, bufA) -> h2 = bufC   [128 -> 128, LN+ReLU]
    vgae_sage_gemm<<<nrt, 256, 0, stream>>>(bufB, bufA, invdeg, Wl1, bl1, Wr1,
                                            g1, b1, bufC, 128, 128, 1);
    // agg(h2) -> bufA (h1 dead, reuse)
    vgae_zero<<<((long)N * 128 + 255) / 256, 256, 0, stream>>>(bufA, (long)N * 128);
    { dim3 blk(128, 2); vgae_agg<<<(E + 1) / 2, blk, 0, stream>>>(bufC, ei, bufA, 128, E); }

    // layer 2: (bufA, bufC) -> out   [128 -> 64, no LN]
    vgae_sage_gemm<<<nrt, 128, 0, stream>>>(bufA, bufC, invdeg, Wl2, bl2, Wr2,
                                            nullptr, nullptr, outp, 128, 64, 0);
  }
}